// SpikingTemporalAttention_1314259992786
// MI455X (gfx1250) — compile-verified
//
#include <hip/hip_runtime.h>
#include <math.h>
#include <stdint.h>

typedef __attribute__((ext_vector_type(2))) float v2f;
typedef __attribute__((ext_vector_type(8))) float v8f;

#define T_     8
#define N_     1024
#define D_     128
#define HEADS_ 4
#define DH_    32
#define WIN_   4
#define E_     32768
#define NEG_BIG  (-3.0e38f)
#define CONSUMED (-3.3e38f)
#define VALID_TH (-1.0e37f)

// ---------------- adjacency bitmask: adj[dst][src bitset], 1024 x 32 words ----
__global__ void adj_kernel(const int* __restrict__ edge, unsigned* __restrict__ adj) {
    int i = blockIdx.x * blockDim.x + threadIdx.x;
    if (i < E_) {
        int src = edge[i];
        int dst = edge[E_ + i];
        atomicOr(&adj[dst * 32 + (src >> 5)], 1u << (src & 31));
    }
}

// ---------------- QKV projection: one wave computes one 16x16 tile, K=128 -----
// A-frag (16x4 f32): lane%16 = M row, K = k0 + 2*(lane/16) + {0,1} in vgpr {0,1}
// B-frag (4x16 f32): lane%16 = N col, same K striping.  B[k][c] = W[c][k].
__global__ __launch_bounds__(32) void proj_kernel(
    const float* __restrict__ X,                       // [T*N, 128]
    const float* __restrict__ Wq, const float* __restrict__ Wk,
    const float* __restrict__ Wv,
    float* __restrict__ Q, float* __restrict__ Kh, float* __restrict__ V)
{
    int bx  = blockIdx.x;
    int mat = bx >> 12;            // 4096 = 512 row-tiles * 8 col-tiles per matrix
    int rem = bx & 4095;
    int row0 = (rem >> 3) * 16;
    int col0 = (rem & 7) * 16;
    const float* W; int ld; float* O;
    if (mat == 0)      { W = Wq; ld = 128; O = Q;  }
    else if (mat == 1) { W = Wk; ld = 136; O = Kh; }   // first 128 cols of Wk
    else               { W = Wv; ld = 128; O = V;  }

    int lane = threadIdx.x;
    int m    = lane & 15;
    int half = lane >> 4;
    const float* xrow = X + (row0 + m) * 128;
    const float* wrow = W + (col0 + m) * ld;

    v8f acc = {};
    #pragma unroll
    for (int kk = 0; kk < 128; kk += 4) {
        int k = kk + 2 * half;
        v2f a; a.x = xrow[k]; a.y = xrow[k + 1];
        v2f b; b.x = wrow[k]; b.y = wrow[k + 1];
        acc = __builtin_amdgcn_wmma_f32_16x16x4_f32(false, a, false, b,
                                                    (short)0, acc, false, false);
    }
    #pragma unroll
    for (int r = 0; r < 8; ++r)
        O[(row0 + 8 * half + r) * 128 + col0 + m] = acc[r];
}

// ---------------- pe_proj[w][c] = sum_j pe[w][j] * Wk[c][128+j]  (5 x 128) ----
__global__ void peproj_kernel(const float* __restrict__ Wk, float* __restrict__ pe_proj) {
    int c = threadIdx.x;
    if (c >= 128) return;
    for (int w = 0; w <= WIN_; ++w) {
        float dt = (float)w;
        float pe[8];
        pe[0] = expf(-dt * 0.25f);     // tau = 4
        pe[1] = expf(-dt * 0.0625f);   // tau = 16
        pe[2] = sinf(dt);  pe[3] = sinf(dt * 0.5f);  pe[4] = sinf(dt * 0.25f);
        pe[5] = cosf(dt);  pe[6] = cosf(dt * 0.5f);  pe[7] = cosf(dt * 0.25f);
        float s = 0.f;
        #pragma unroll
        for (int j = 0; j < 8; ++j) s += pe[j] * Wk[c * 136 + 128 + j];
        pe_proj[w * 128 + c] = s;
    }
}

// ---------------- gate_log = log(clip(S,0,1)+1e-6) ---------------------------
__global__ void gate_kernel(const float* __restrict__ S, float* __restrict__ gl, int n) {
    int i = blockIdx.x * blockDim.x + threadIdx.x;
    if (i < n) {
        float s = fminf(fmaxf(S[i], 0.f), 1.f);
        gl[i] = logf(s + 1e-6f);
    }
}

// ---------------- fused score GEMM + masked top-16 softmax + V aggregation ----
// block = 256 threads (8 waves) handles (t, h, 16 dst rows).
// K rows for the *next* 256-candidate chunk are prefetched into LDS with
// global_load_async_to_lds_b128 (ASYNCcnt double buffering): each wave loads
// exactly the 32 rows its own two 16x16 score tiles consume, so a per-wave
// s_wait_asynccnt suffices (no block barrier on the K path).
__global__ __launch_bounds__(256) void attn_kernel(
    const float* __restrict__ Q, const float* __restrict__ Kh,
    const float* __restrict__ V, const float* __restrict__ pe_proj,
    const float* __restrict__ gl, const unsigned* __restrict__ adj,
    float* __restrict__ out)
{
    __shared__ __align__(16) float Kst[2][256][36];  // padded: 36m mod 64 distinct
    __shared__ float    Qs[16][36];                  // Q tile, pre-scaled
    __shared__ float    peS[32];
    __shared__ float    glS[256];
    __shared__ unsigned adjW[16][8];
    __shared__ float    chunkS[16][260];             // half-waves hit disjoint banks
    __shared__ float    candV[16][256];
    __shared__ int      candI[16][256];
    __shared__ float    pS[16][16];
    __shared__ int      iS[16][16];
    __shared__ float    redV[16][16];
    __shared__ int      redQ[16][16];

    int bx = blockIdx.x;
    int rt = bx & 63;  bx >>= 6;
    int h  = bx & 3;
    int t  = bx >> 2;
    int row0 = rt * 16;
    int tid  = threadIdx.x;
    int lane = tid & 31, wave = tid >> 5;
    int row  = tid >> 4, off  = tid & 15;

    const float SCALE = 0.17677669529663687f;   // 32^-0.5
    for (int i = tid; i < 16 * 32; i += 256) {
        int r = i >> 5, d = i & 31;
        Qs[r][d] = Q[(t * N_ + row0 + r) * D_ + h * DH_ + d] * SCALE;
    }

    unsigned kstBase = (unsigned)(uintptr_t)(&Kst[0][0][0]);
    // async-stage the K head-slice rows of chunk cc into buffer buf.
    // thread tid owns within-chunk row tid -> 8 x b128 (32 floats).
    auto issueK = [&](int cc, int buf) {
        int w2 = cc >> 2, tp2 = t - w2, mb02 = (cc & 3) * 256;
        const float* g = Kh + (size_t)(tp2 * N_ + mb02 + tid) * D_ + h * DH_;
        unsigned l = kstBase + (unsigned)((buf * 256 + tid) * 36 * 4);
        #pragma unroll
        for (int kg = 0; kg < 8; ++kg) {
            unsigned long long ga = (unsigned long long)(uintptr_t)(g + kg * 4);
            unsigned la = l + (unsigned)(kg * 16);
            asm volatile("global_load_async_to_lds_b128 %0, %1, off"
                         :: "v"(la), "v"(ga) : "memory");
        }
    };

    float tv[16]; int ti[16];
    #pragma unroll
    for (int k = 0; k < 16; ++k) { tv[k] = NEG_BIG; ti[k] = 0; }
    float minv = NEG_BIG; int minpos = 0;

    issueK(0, 0);                          // prefetch first chunk
    __syncthreads();

    for (int c = 0; c < 20; ++c) {
        int w = c >> 2;
        if (w > t) break;                  // chunks ordered by w; uniform exit
        int mb0 = (c & 3) * 256;           // global m base of chunk
        int tp  = t - w;
        int buf = c & 1;

        if (tid < 32) peS[tid] = pe_proj[w * 128 + h * DH_ + tid];
        glS[tid] = gl[tp * N_ + mb0 + tid];
        if (tid < 128)
            adjW[tid >> 3][tid & 7] = adj[(row0 + (tid >> 3)) * 32 + (mb0 >> 5) + (tid & 7)];

        bool pf = (c < 19) && (((c + 1) >> 2) <= t);
        if (pf) {
            issueK(c + 1, (c + 1) & 1);
            asm volatile("s_wait_asynccnt 0x8" ::: "memory");  // this chunk done
        } else {
            asm volatile("s_wait_asynccnt 0x0" ::: "memory");
        }
        __syncthreads();

        float relb = -log1pf((float)w);
        int mcol = lane & 15, half = lane >> 4;

        #pragma unroll
        for (int sub = 0; sub < 2; ++sub) {
            int mb = (wave * 2 + sub) * 16;       // within-chunk col base
            v8f acc = {};
            #pragma unroll
            for (int kk = 0; kk < 32; kk += 4) {
                int k = kk + 2 * half;
                int ml = mb + mcol;
                v2f a; a.x = Qs[mcol][k];               a.y = Qs[mcol][k + 1];
                v2f b; b.x = Kst[buf][ml][k] + peS[k];  b.y = Kst[buf][ml][k + 1] + peS[k + 1];
                acc = __builtin_amdgcn_wmma_f32_16x16x4_f32(false, a, false, b,
                                                            (short)0, acc, false, false);
            }
            #pragma unroll
            for (int r = 0; r < 8; ++r) {
                int lr = 8 * half + r;
                int cc = mb + mcol;
                bool ok = (adjW[lr][cc >> 5] >> (cc & 31)) & 1u;
                chunkS[lr][cc] = ok ? (acc[r] + relb + glS[cc]) : NEG_BIG;
            }
        }
        __syncthreads();

        // online top-16: 16 threads/row, each scans 16 columns of this chunk
        int idxbase = w * N_ + mb0;        // flattened candidate index = w*N + m
        for (int j = 0; j < 16; ++j) {
            int cc = j * 16 + off;
            float val = chunkS[row][cc];
            if (val > VALID_TH && val > minv) {
                tv[minpos] = val; ti[minpos] = idxbase + cc;
                minv = tv[0]; minpos = 0;
                #pragma unroll
                for (int k = 1; k < 16; ++k)
                    if (tv[k] < minv) { minv = tv[k]; minpos = k; }
            }
        }
        __syncthreads();
    }

    #pragma unroll
    for (int k = 0; k < 16; ++k) { candV[row][off * 16 + k] = tv[k];
                                   candI[row][off * 16 + k] = ti[k]; }
    __syncthreads();

    // merge 256 -> top-16: 16 rounds, parallel across the 16 threads of a row
    #pragma unroll
    for (int k = 0; k < 16; ++k) {
        float best = CONSUMED; int bq = off * 16;
        #pragma unroll
        for (int q = 0; q < 16; ++q) {
            float v = candV[row][off * 16 + q];
            if (v > best) { best = v; bq = off * 16 + q; }
        }
        redV[row][off] = best; redQ[row][off] = bq;
        __syncthreads();
        if (off == 0) {
            float gb = redV[row][0]; int go = 0;
            #pragma unroll
            for (int o = 1; o < 16; ++o)
                if (redV[row][o] > gb) { gb = redV[row][o]; go = o; }
            int q = redQ[row][go];
            pS[row][k] = gb; iS[row][k] = candI[row][q];
            candV[row][q] = CONSUMED;
        }
        __syncthreads();
    }

    // softmax over the kept 16 (all masked -> zero row, matching reference)
    if (off == 0) {
        float maxv = pS[row][0];
        if (maxv <= VALID_TH) {
            for (int k = 0; k < 16; ++k) { pS[row][k] = 0.f; iS[row][k] = 0; }
        } else {
            float sum = 0.f; float e[16];
            #pragma unroll
            for (int k = 0; k < 16; ++k) {
                float v = pS[row][k];
                e[k] = (v > VALID_TH) ? expf(v - maxv) : 0.f;
                sum += e[k];
            }
            float inv = 1.f / sum;
            #pragma unroll
            for (int k = 0; k < 16; ++k) pS[row][k] = e[k] * inv;
        }
    }
    __syncthreads();

    // aggregate V: 16 threads/row, 2 dims each
    int d0 = off * 2;
    float a0 = 0.f, a1 = 0.f;
    for (int k = 0; k < 16; ++k) {
        float p = pS[row][k];
        if (p > 0.f) {
            int idx = iS[row][k];
            int wv = idx >> 10, m = idx & 1023;
            const float* vp = V + ((t - wv) * N_ + m) * D_ + h * DH_ + d0;
            a0 += p * vp[0];
            a1 += p * vp[1];
        }
    }
    int n = row0 + row;
    out[(t * N_ + n) * D_ + h * DH_ + d0]     = a0;
    out[(t * N_ + n) * D_ + h * DH_ + d0 + 1] = a1;
}

extern "C" void kernel_launch(void* const* d_in, const int* in_sizes, int n_in,
                              void* d_out, int out_size, void* d_ws, size_t ws_size,
                              hipStream_t stream) {
    const float* H    = (const float*)d_in[0];   // [8,1024,128]
    const float* S    = (const float*)d_in[1];   // [8,1024]
    const float* Wq   = (const float*)d_in[2];   // [128,128]
    const float* Wk   = (const float*)d_in[3];   // [128,136]
    const float* Wv   = (const float*)d_in[4];   // [128,128]
    const int*   edge = (const int*)d_in[5];     // [2,32768]
    float* out = (float*)d_out;

    float* ws   = (float*)d_ws;
    float* Q    = ws;                       // 1M floats
    float* Kh   = ws + 1048576;             // 1M floats
    float* Vp   = ws + 2 * 1048576;         // 1M floats
    float* pe   = ws + 3 * 1048576;         // 640 (pad 1024)
    float* gl   = pe + 1024;                // 8192
    unsigned* adj = (unsigned*)(gl + 8192); // 32768 words

    hipMemsetAsync(adj, 0, N_ * 32 * sizeof(unsigned), stream);
    adj_kernel<<<E_ / 256, 256, 0, stream>>>(edge, adj);
    proj_kernel<<<3 * 512 * 8, 32, 0, stream>>>(H, Wq, Wk, Wv, Q, Kh, Vp);
    peproj_kernel<<<1, 128, 0, stream>>>(Wk, pe);
    gate_kernel<<<(T_ * N_ + 255) / 256, 256, 0, stream>>>(S, gl, T_ * N_);
    attn_kernel<<<T_ * HEADS_ * 64, 256, 0, stream>>>(Q, Kh, Vp, pe, gl, adj, out);
}